// LatentTimeSeriesMixtureWeights_80994493268007
// MI455X (gfx1250) — compile-verified
//
#include <hip/hip_runtime.h>
#include <math.h>

#define K_LAT 100
#define T_N   10
#define S_N   10
#define RMAXN 5000
#define O_N   128
#define H_N   50
#define NSEG  8

typedef float v2f __attribute__((ext_vector_type(2)));
typedef float v8f __attribute__((ext_vector_type(8)));

__device__ __forceinline__ float softplus_f(float x) {
    return fmaxf(x, 0.f) + log1pf(expf(-fabsf(x)));
}

__device__ __forceinline__ v8f wmma4(v2f a, v2f b, v8f c) {
    return __builtin_amdgcn_wmma_f32_16x16x4_f32(false, a, false, b, (short)0, c,
                                                 false, false);
}

__device__ __forceinline__ void wait_async0() {
#if __has_builtin(__builtin_amdgcn_s_wait_asynccnt)
    __builtin_amdgcn_s_wait_asynccnt(0);
#else
    asm volatile("s_wait_asynccnt 0x0" ::: "memory");
#endif
}

// Issue one 128-bit async global->LDS transfer (CDNA5 GLOBAL_LOAD_ASYNC_TO_LDS).
__device__ __forceinline__ void async_copy_b128(uint32_t lds_byte_addr,
                                                uint64_t global_byte_addr) {
    asm volatile("global_load_async_to_lds_b128 %0, %1, off"
                 :: "v"(lds_byte_addr), "v"(global_byte_addr) : "memory");
}

// ---------------------------------------------------------------------------
// Kernel A: per-(t,s) ragged MLP encode + masked sum pool (segmented).
// grid = (T*S, NSEG), 128 threads (4 waves). Wave w owns output columns
// [16w,16w+16). Double-buffered async A-tile pipeline; B-fragments hoisted
// into registers. Partial column sums atomically accumulated into enc_sum.
// ---------------------------------------------------------------------------
__global__ void __launch_bounds__(128)
encode_kernel(const float* __restrict__ reads, const int* __restrict__ counts,
              const float* __restrict__ W1, const float* __restrict__ b1,
              const float* __restrict__ W2, const float* __restrict__ b2,
              float* __restrict__ enc_sum) {
    __shared__ float sA[2][16 * O_N];   // double-buffered A tiles (2 x 8KB)
    __shared__ float sH1[16 * 64];      // layer-1 activation tile
    __shared__ float sPool[128];

    const int blk  = blockIdx.x;          // t*S + s
    const int seg  = blockIdx.y;
    const int tid  = threadIdx.x;
    const int wave = tid >> 5;
    const int lane = tid & 31;
    const int nloc = lane & 15;
    const int hi8  = (lane >> 4) << 3;    // C/D row offset: 0 or 8
    const int koff = (lane >> 4) << 1;    // A/B K offset: 0 or 2
    const int n    = wave * 16 + nloc;    // output column of this lane
    const int count = counts[blk];
    const float* Abase = reads + (size_t)blk * RMAXN * O_N;
    const uint64_t gbase = (uint64_t)(uintptr_t)Abase;
    const uint32_t lbase0 = (uint32_t)(uintptr_t)&sA[0][0];
    const uint32_t lbase1 = (uint32_t)(uintptr_t)&sA[1][0];

    // ---- hoist B fragments into registers (L2-resident weights) ----
    const bool okn = (n < H_N);
    v2f b1F[32];
    #pragma unroll
    for (int kk = 0; kk < 32; ++kk) {
        int kb = (kk << 2) + koff;
        b1F[kk].x = okn ? W1[kb * H_N + n]       : 0.f;
        b1F[kk].y = okn ? W1[(kb + 1) * H_N + n] : 0.f;
    }
    v2f b2F[16];
    #pragma unroll
    for (int kk = 0; kk < 16; ++kk) {
        int kb = (kk << 2) + koff;
        b2F[kk].x = (kb < H_N && okn)     ? W2[kb * H_N + n]       : 0.f;
        b2F[kk].y = (kb + 1 < H_N && okn) ? W2[(kb + 1) * H_N + n] : 0.f;
    }
    const float bv1 = okn ? b1[n] : 0.f;
    const float bv2 = okn ? b2[n] : 0.f;

    const int mTiles = (count + 15) >> 4;

    // issue a 16x128 tile: 512 x 16B chunks, 4 per thread
    auto issue_tile = [&](uint32_t lbase, int mt) {
        const int m0 = mt << 4;
        #pragma unroll
        for (int j = 0; j < 4; ++j) {
            int c  = tid + j * 128;        // chunk id 0..511
            int r  = c >> 5;               // tile row (512B per row)
            int co = c & 31;               // 16B chunk within row
            int row = m0 + r;
            row = (row < count) ? row : (count - 1);   // clamp: stays in-bounds
            uint64_t ga = gbase + ((uint64_t)(uint32_t)row << 9) + ((uint32_t)co << 4);
            uint32_t la = lbase + (uint32_t)((r << 9) + (co << 4));
            async_copy_b128(la, ga);
        }
    };

    float pool = 0.f;
    int mt = seg;
    if (mt < mTiles) issue_tile(lbase0, mt);
    for (int it = 0; mt < mTiles; mt += NSEG, ++it) {
        const int cur = it & 1;
        const float* At = sA[cur];
        wait_async0();
        __syncthreads();                   // tile mt resident in LDS
        if (mt + NSEG < mTiles) issue_tile(cur ? lbase0 : lbase1, mt + NSEG);

        const int m0 = mt << 4;
        // ---- layer 1: [16x128] x [128x16], 32 WMMA, B in registers ----
        v8f c1;
        #pragma unroll
        for (int v = 0; v < 8; ++v) c1[v] = bv1;
        #pragma unroll
        for (int kk = 0; kk < 32; ++kk) {
            const int kb = (kk << 2) + koff;
            v2f a;
            a.x = At[nloc * O_N + kb];
            a.y = At[nloc * O_N + kb + 1];
            c1 = wmma4(a, b1F[kk], c1);
        }
        #pragma unroll
        for (int v = 0; v < 8; ++v)
            sH1[(v + hi8) * 64 + n] = softplus_f(c1[v]);
        __syncthreads();

        // ---- layer 2: [16x64] x [64x16], 16 WMMA ----
        v8f c2;
        #pragma unroll
        for (int v = 0; v < 8; ++v) c2[v] = bv2;
        #pragma unroll
        for (int kk = 0; kk < 16; ++kk) {
            const int kb = (kk << 2) + koff;
            v2f a;
            a.x = sH1[nloc * 64 + kb];
            a.y = sH1[nloc * 64 + kb + 1];
            c2 = wmma4(a, b2F[kk], c2);
        }
        #pragma unroll
        for (int v = 0; v < 8; ++v) {
            int row = m0 + v + hi8;
            float hv = softplus_f(c2[v]);
            pool += (row < count) ? hv : 0.f;
        }
        __syncthreads();
    }

    sPool[tid] = pool;
    __syncthreads();
    if (lane < 16) {
        float tot = sPool[wave * 32 + lane] + sPool[wave * 32 + lane + 16];
        if (n < H_N && tot != 0.f) atomicAdd(&enc_sum[blk * 64 + n], tot);
        else if (n < H_N) atomicAdd(&enc_sum[blk * 64 + n], tot);
    }
}

// zero enc_sum and KL slot
__global__ void init_kernel(float* __restrict__ enc_sum, float* __restrict__ kl) {
    int i = blockIdx.x * 256 + threadIdx.x;
    if (i < (T_N * S_N) * 64) enc_sum[i] = 0.f;
    if (i == 0) kl[0] = 0.f;
}

// enc = enc_sum / count, zero padded columns
__global__ void finalize_kernel(float* __restrict__ enc_sum,
                                const int* __restrict__ counts) {
    int i = blockIdx.x * 256 + threadIdx.x;
    if (i < (T_N * S_N) * 64) {
        int col = i & 63, row = i >> 6;
        float v = (col < H_N) ? enc_sum[i] / (float)counts[row] : 0.f;
        enc_sum[i] = v;
    }
}

// ---------------------------------------------------------------------------
// Kernel B: mu/var heads + reparameterized sample. enc[100x64] x W[64x112pad].
// ---------------------------------------------------------------------------
__global__ void __launch_bounds__(32)
latent_kernel(const float* __restrict__ enc,
              const float* __restrict__ Wmu, const float* __restrict__ bmu,
              const float* __restrict__ Wvar, const float* __restrict__ bvar,
              const float* __restrict__ eps,
              float* __restrict__ xout, float* __restrict__ mu_ws,
              float* __restrict__ var_ws) {
    const int lane = threadIdx.x;
    const int nloc = lane & 15;
    const int hi8  = (lane >> 4) << 3;
    const int koff = (lane >> 4) << 1;
    const int n    = blockIdx.x * 16 + nloc;   // latent index k

    v2f bmuF[16], bvarF[16];
    #pragma unroll
    for (int kk = 0; kk < 16; ++kk) {
        int kb = (kk << 2) + koff;
        bool okn = (n < K_LAT);
        bmuF[kk].x  = (kb < H_N && okn)     ? Wmu[kb * K_LAT + n]        : 0.f;
        bmuF[kk].y  = (kb + 1 < H_N && okn) ? Wmu[(kb + 1) * K_LAT + n]  : 0.f;
        bvarF[kk].x = (kb < H_N && okn)     ? Wvar[kb * K_LAT + n]       : 0.f;
        bvarF[kk].y = (kb + 1 < H_N && okn) ? Wvar[(kb + 1) * K_LAT + n] : 0.f;
    }
    const float bmuv  = (n < K_LAT) ? bmu[n]  : 0.f;
    const float bvarv = (n < K_LAT) ? bvar[n] : 0.f;

    for (int mt = 0; mt < 7; ++mt) {
        v8f cmu, cvar;
        #pragma unroll
        for (int v = 0; v < 8; ++v) { cmu[v] = bmuv; cvar[v] = bvarv; }
        #pragma unroll
        for (int kk = 0; kk < 16; ++kk) {
            int kb  = (kk << 2) + koff;
            int row = mt * 16 + nloc;
            v2f a;
            a.x = (row < T_N * S_N) ? enc[row * 64 + kb]     : 0.f;
            a.y = (row < T_N * S_N) ? enc[row * 64 + kb + 1] : 0.f;
            cmu  = wmma4(a, bmuF[kk],  cmu);
            cvar = wmma4(a, bvarF[kk], cvar);
        }
        #pragma unroll
        for (int v = 0; v < 8; ++v) {
            int row = mt * 16 + v + hi8;      // t*S + s
            if (row < T_N * S_N && n < K_LAT) {
                float m  = cmu[v];
                float va = expf(cvar[v]);
                float x  = m + sqrtf(va) * eps[n * (T_N * S_N) + row];
                xout[n * (T_N * S_N) + row] = x;
                mu_ws[row * K_LAT + n]  = m;
                var_ws[row * K_LAT + n] = va;
            }
        }
    }
}

// ---------------------------------------------------------------------------
// Kernel C: KL reduction over all (k,t,s).
// ---------------------------------------------------------------------------
__global__ void __launch_bounds__(256)
kl_kernel(const float* __restrict__ xout, const float* __restrict__ mu_ws,
          const float* __restrict__ var_ws, const float* __restrict__ delta,
          const float* __restrict__ c_ind, const float* __restrict__ var_process,
          float* __restrict__ kl) {
    __shared__ float red[256];
    const int idx = blockIdx.x * 256 + threadIdx.x;   // k*100 + t*10 + s
    float contrib = 0.f;
    if (idx < K_LAT * T_N * S_N) {
        const int k   = idx / (T_N * S_N);
        const int rem = idx - k * (T_N * S_N);
        const int t   = rem / S_N;
        const int s   = rem - t * S_N;
        const float m  = mu_ws[rem * K_LAT + k];
        const float va = var_ws[rem * K_LAT + k];
        if (t == 0) {
            contrib = 0.5f * (va + m * m - 1.f - logf(va));
        } else {
            // PERTURBED=[0,0,0,1,1,1,0,0,0,0] -> idx {2,2,0,1,2,2,2,2,2}, P=3
            const int   pi   = (t == 3) ? 0 : ((t == 4) ? 1 : 2);
            const float sign = (t >= 3 && t <= 5) ? 1.f : -1.f;
            const float coef = c_ind[k * 3 + pi] * delta[k * 3 + pi] * sign;
            const float x    = xout[idx];
            const float xp   = xout[k * (T_N * S_N) + (t - 1) * S_N + s];
            const float eta  = xp + coef;
            const float vp   = var_process[s];          // DT == 1
            const float dq   = x - m;
            const float dp   = x - eta;
            const float logq = -0.5f * logf(va) - 0.5f * dq * dq / va;
            const float logp = -0.5f * logf(vp) - 0.5f * dp * dp / vp;
            contrib = logq - logp;
        }
    }
    red[threadIdx.x] = contrib;
    __syncthreads();
    for (int off = 128; off > 0; off >>= 1) {
        if (threadIdx.x < off) red[threadIdx.x] += red[threadIdx.x + off];
        __syncthreads();
    }
    if (threadIdx.x == 0) atomicAdd(kl, red[0]);
}

// ---------------------------------------------------------------------------
extern "C" void kernel_launch(void* const* d_in, const int* in_sizes, int n_in,
                              void* d_out, int out_size, void* d_ws, size_t ws_size,
                              hipStream_t stream) {
    const float* reads  = (const float*)d_in[0];
    const int*   counts = (const int*)  d_in[1];
    const float* W1     = (const float*)d_in[2];
    const float* b1     = (const float*)d_in[3];
    const float* W2     = (const float*)d_in[4];
    const float* b2     = (const float*)d_in[5];
    const float* Wmu    = (const float*)d_in[6];
    const float* bmu    = (const float*)d_in[7];
    const float* Wvar   = (const float*)d_in[8];
    const float* bvar   = (const float*)d_in[9];
    const float* eps    = (const float*)d_in[10];
    const float* delta  = (const float*)d_in[11];
    const float* c_ind  = (const float*)d_in[12];
    const float* var_p  = (const float*)d_in[13];

    float* out  = (float*)d_out;
    float* xout = out;                                   // [K,T,S]
    float* kl   = out + K_LAT * T_N * S_N;

    float* ws     = (float*)d_ws;
    float* enc    = ws;                                  // 100*64 (sum -> mean)
    float* mu_ws  = ws + (T_N * S_N) * 64;               // 100*100
    float* var_ws = mu_ws + (T_N * S_N) * K_LAT;         // 100*100

    init_kernel<<<((T_N * S_N) * 64 + 255) / 256, 256, 0, stream>>>(enc, kl);
    encode_kernel<<<dim3(T_N * S_N, NSEG), 128, 0, stream>>>(
        reads, counts, W1, b1, W2, b2, enc);
    finalize_kernel<<<((T_N * S_N) * 64 + 255) / 256, 256, 0, stream>>>(enc, counts);
    latent_kernel<<<7, 32, 0, stream>>>(enc, Wmu, bmu, Wvar, bvar, eps,
                                        xout, mu_ws, var_ws);
    kl_kernel<<<(K_LAT * T_N * S_N + 255) / 256, 256, 0, stream>>>(
        xout, mu_ws, var_ws, delta, c_ind, var_p, kl);
}